// NeuSRenderer_30820685316807
// MI455X (gfx1250) — compile-verified
//
#include <hip/hip_runtime.h>
#include <stdint.h>

// NeuS hierarchical sampling on MI455X (gfx1250).
// One ray per wave32, one wave per workgroup. Per-ray state in LDS.
// Interval math + scans fully register-resident (contiguous 8 intervals/lane,
// shfl_up for the cross-lane shifted-min and Kogge-Stone scans); LDS only for
// the data that genuinely needs random access (z, sdf, cdf, new samples).
// Final row emitted via the CDNA5 async store-from-LDS path (ASYNCcnt).

constexpr int kNumSteps = 128;   // NUM_STEPS
constexpr int kNImp     = 32;    // UPSAMPLE_STEPS / UP_SAMPLE_ITERS
constexpr int kIters    = 4;     // UP_SAMPLE_ITERS
constexpr int kMaxK     = kNumSteps + kIters * kNImp;  // 256

__device__ __forceinline__ float fsigmoid(float x) {
  return 1.0f / (1.0f + __expf(-x));
}

__global__ __launch_bounds__(32) void neus_upsample_kernel(
    const float* __restrict__ rays_o,
    const float* __restrict__ rays_d,
    float* __restrict__ out, int nrays) {
  const int ray  = blockIdx.x;
  if (ray >= nrays) return;
  const int lane = threadIdx.x;  // wave32: lane == threadIdx.x

  __shared__ __align__(16) float zA[kMaxK];
  __shared__ __align__(16) float zB[kMaxK];
  __shared__ __align__(16) float sA[kMaxK];
  __shared__ __align__(16) float sB[kMaxK];
  __shared__ float cdf[kMaxK];
  __shared__ float nzv[kNImp];    // new z samples (sorted)
  __shared__ float nsv[kNImp];    // new sdf samples

  // ---- load ray, normalize direction (wave-uniform -> SALU/SMEM) ----
  float ox = rays_o[3 * ray + 0], oy = rays_o[3 * ray + 1], oz = rays_o[3 * ray + 2];
  float dx = rays_d[3 * ray + 0], dy = rays_d[3 * ray + 1], dz = rays_d[3 * ray + 2];
  {
    float inv = 1.0f / (sqrtf(dx * dx + dy * dy + dz * dz) + 1e-8f);
    dx *= inv; dy *= inv; dz *= inv;
  }

  auto sdf_at = [&](float t) -> float {
    float px = fminf(fmaxf(fmaf(dx, t, ox), -1.0f), 1.0f);
    float py = fminf(fmaxf(fmaf(dy, t, oy), -1.0f), 1.0f);
    float pz = fminf(fmaxf(fmaf(dz, t, oz), -1.0f), 1.0f);
    return sqrtf(px * px + py * py + pz * pz) - 0.5f;
  };

  float* zc = zA; float* zn = zB;   // ping-pong (current / next)
  float* sc = sA; float* sn = sB;

  // ---- initial uniform samples ----
  for (int i = lane; i < kNumSteps; i += 32) {
    float t = fmaf(1.8f, (float)i * (1.0f / (float)(kNumSteps - 1)), 0.2f);
    zc[i] = t;
    sc[i] = sdf_at(t);
  }
  __syncthreads();

  int K = kNumSteps;
  for (int it = 0; it < kIters; ++it) {
    const float inv_s = 64.0f * (float)(1 << it);
    const int   M     = K - 1;            // #intervals
    const bool  last  = (it + 1 == kIters);
    const int   base  = lane * 8;         // this lane owns intervals [base, base+8)

    // ---- interval math, register-resident ----
    float zloc[9], sloc[9];
#pragma unroll
    for (int j = 0; j < 9; ++j) {
      int idx = base + j;
      int ii  = (idx < K) ? idx : (K - 1);   // clamped; padded slots unused
      zloc[j] = zc[ii];
      sloc[j] = sc[ii];
    }
    float craw[8];
#pragma unroll
    for (int j = 0; j < 8; ++j) {
      bool ok = (base + j) < M;
      craw[j] = ok ? (sloc[j + 1] - sloc[j]) / (zloc[j + 1] - zloc[j] + 1e-5f)
                   : 0.0f;
    }
    // shifted prev_cos for slot 0 comes from the neighbor lane's slot 7
    float cprev0 = __shfl_up(craw[7], 1u, 32);
    if (lane == 0) cprev0 = 0.0f;          // reference: prev_cos[0] = 0

    float av[8];
#pragma unroll
    for (int j = 0; j < 8; ++j) {
      float pc = (j > 0) ? craw[j - 1] : cprev0;
      float c  = fminf(pc, craw[j]);
      c = fminf(fmaxf(c, -1000.0f), 0.0f);
      float mid  = 0.5f * (sloc[j] + sloc[j + 1]);
      float dist = zloc[j + 1] - zloc[j];
      float pe   = fmaf(-c * 0.5f, dist, mid);
      float ne   = fmaf( c * 0.5f, dist, mid);
      float pcdf = fsigmoid(pe * inv_s);
      float ncdf = fsigmoid(ne * inv_s);
      bool  ok   = (base + j) < M;
      av[j] = ok ? (pcdf - ncdf + 1e-5f) / (pcdf + 1e-5f) : 0.0f;
    }

    // ---- cumprod (transmittance) + cumsum (CDF) via Kogge-Stone ----
    // lane-local product of (1 - alpha + 1e-7), identity-padded
    float prod = 1.0f;
#pragma unroll
    for (int j = 0; j < 8; ++j) {
      bool ok = (base + j) < M;
      prod *= ok ? (1.0f - av[j] + 1e-7f) : 1.0f;
    }
    float inclP = prod;
    for (int d = 1; d < 32; d <<= 1) {
      float t = __shfl_up(inclP, (unsigned)d, 32);
      if (lane >= d) inclP *= t;
    }
    float exProd = __shfl_up(inclP, 1u, 32);
    if (lane == 0) exProd = 1.0f;

    // weights (+1e-5) with running transmittance; local exclusive sums
    float es[8], sum = 0.0f, tr = exProd;
#pragma unroll
    for (int j = 0; j < 8; ++j) {
      bool  ok = (base + j) < M;
      float w2 = ok ? fmaf(av[j], tr, 1e-5f) : 0.0f;
      es[j] = sum; sum += w2;
      tr *= ok ? (1.0f - av[j] + 1e-7f) : 1.0f;
    }
    float inclS = sum;
    for (int d = 1; d < 32; d <<= 1) {
      float t = __shfl_up(inclS, (unsigned)d, 32);
      if (lane >= d) inclS += t;
    }
    float total = __shfl(inclS, 31, 32);
    float exSum = __shfl_up(inclS, 1u, 32);
    if (lane == 0) exSum = 0.0f;
    float invTot = 1.0f / total;
#pragma unroll
    for (int j = 0; j < 8; ++j) {
      int k = base + j;
      if (k < K) cdf[k] = (exSum + es[j]) * invTot;  // cdf[0]=0 .. cdf[K-1]=1
    }
    __syncthreads();

    // ---- inverse-CDF sampling: one new sample per lane ----
    {
      float u  = ((float)lane + 0.5f) * (1.0f / (float)kNImp);
      int lo = 0, hi = K;                 // searchsorted(cdf, u, side='right')
      while (lo < hi) {
        int mid = (lo + hi) >> 1;
        if (cdf[mid] <= u) lo = mid + 1; else hi = mid;
      }
      int below = lo - 1; if (below < 0) below = 0;
      int above = lo;     if (above > K - 1) above = K - 1;
      float cb = cdf[below], ca = cdf[above];
      float bb = zc[below],  ba = zc[above];
      float den = ca - cb;
      den = (den < 1e-5f) ? 1.0f : den;
      float t  = (u - cb) / den;
      float zv = fmaf(t, ba - bb, bb);
      nzv[lane] = zv;
      if (!last) nsv[lane] = sdf_at(zv);
    }
    __syncthreads();

    // ---- stable merge (matches stable argsort of [old, new]) ----
    for (int i = lane; i < K; i += 32) {
      float v = zc[i];
      int lo = 0, hi = kNImp;             // count of new strictly < v
      while (lo < hi) {
        int mid = (lo + hi) >> 1;
        if (nzv[mid] < v) lo = mid + 1; else hi = mid;
      }
      zn[i + lo] = v;
      if (!last) sn[i + lo] = sc[i];
    }
    {
      float v = nzv[lane];
      int lo = 0, hi = K;                 // count of old <= v
      while (lo < hi) {
        int mid = (lo + hi) >> 1;
        if (zc[mid] <= v) lo = mid + 1; else hi = mid;
      }
      zn[lo + lane] = v;
      if (!last) sn[lo + lane] = nsv[lane];
    }
    __syncthreads();

    { float* t = zc; zc = zn; zn = t; }
    { float* t = sc; sc = sn; sn = t; }
    K += kNImp;
  }

  // ---- emit final 256-sample row (1 KB) ----
  float* gout = out + (size_t)ray * (size_t)kMaxK;
#if defined(__gfx1250__)
  {
    // Result already lives in LDS: use CDNA5 async store-from-LDS
    // (2 x b128 per lane = 1 KB per wave), tracked with ASYNCcnt.
    asm volatile("s_wait_dscnt 0x0" ::: "memory");
    unsigned long long g0 = (unsigned long long)(uintptr_t)(gout + lane * 4);
    unsigned long long g1 = g0 + 512ull;
    unsigned int l0 = (unsigned int)(uintptr_t)(zc + lane * 4);  // low 32 bits = LDS offset
    unsigned int l1 = l0 + 512u;
    asm volatile(
        "global_store_async_from_lds_b128 %0, %2, off\n\t"
        "global_store_async_from_lds_b128 %1, %3, off"
        :: "v"(g0), "v"(g1), "v"(l0), "v"(l1)
        : "memory");
    asm volatile("s_wait_asynccnt 0x0" ::: "memory");
  }
#else
  {
    float4*       o4 = reinterpret_cast<float4*>(gout);
    const float4* s4 = reinterpret_cast<const float4*>(zc);
    o4[lane]      = s4[lane];
    o4[lane + 32] = s4[lane + 32];
  }
#endif
}

extern "C" void kernel_launch(void* const* d_in, const int* in_sizes, int n_in,
                              void* d_out, int out_size, void* d_ws, size_t ws_size,
                              hipStream_t stream) {
  (void)n_in; (void)out_size; (void)d_ws; (void)ws_size;
  const float* rays_o = (const float*)d_in[0];
  const float* rays_d = (const float*)d_in[1];
  float* out = (float*)d_out;
  const int nrays = in_sizes[0] / 3;
  neus_upsample_kernel<<<dim3(nrays), dim3(32), 0, stream>>>(rays_o, rays_d, out, nrays);
}